// ProjectionLayer_75685913690509
// MI455X (gfx1250) — compile-verified
//
#include <hip/hip_runtime.h>
#include <hip/hip_bf16.h>

typedef float v2f __attribute__((ext_vector_type(2)));
typedef float v8f __attribute__((ext_vector_type(8)));

#define BATCH 512
#define IN_C  2048
#define TT    196
#define OUT_C 512

// ---------------------------------------------------------------------------
// Kernel 1: mean-pool over t.  features [BATCH, IN_C, TT] -> gpool [BATCH*IN_C]
// One wave32 per row. 196 f32 = 49 float4 (row start is 16B aligned: 784%16==0).
// This kernel carries ~822MB of HBM traffic -> it IS the runtime (~35us @23.3TB/s).
// ---------------------------------------------------------------------------
__global__ void pool_mean_kernel(const float* __restrict__ feat,
                                 float* __restrict__ gpool) {
    const int lane = threadIdx.x & 31;
    const int wave = threadIdx.x >> 5;
    const long long row = (long long)blockIdx.x * (blockDim.x >> 5) + wave;

    const float4* p = (const float4*)(feat + row * (long long)TT);
    float4 v = p[lane];                       // f4 idx 0..31
    float s = v.x + v.y + v.z + v.w;
    if (lane < 17) {                          // f4 idx 32..48
        float4 w = p[32 + lane];
        s += w.x + w.y + w.z + w.w;
    }
    // wave32 butterfly reduction
    #pragma unroll
    for (int off = 16; off > 0; off >>= 1)
        s += __shfl_xor(s, off, 32);
    if (lane == 0)
        gpool[row] = s * (1.0f / (float)TT);
}

// ---------------------------------------------------------------------------
// f32 WMMA fragment conventions (CDNA5 ISA, 32-bit layouts, wave32):
//   A 16x4 (MxK): lane L: m = L&15, half = L>>4; VGPR0=A[m][2*half], VGPR1=A[m][2*half+1]
//   B 4x16 (KxN): lane L: n = L&15;              VGPR0=B[2*half][n], VGPR1=B[2*half+1][n]
//   C/D 16x16:    lane L, VGPR v: row = v + 8*half, col = L&15
// Both GEMMs below are D = X * Y^T with X,Y row-major and K contiguous, so both
// A and B fragments are single float2 loads from consecutive K elements.
// ---------------------------------------------------------------------------

// Kernel 2: gproj[m][n] = sum_k gpool[m][k] * W[n][k] + bias[n]
// M=N=512, K=2048. One wave per 16x16 tile; 4 waves (m-adjacent tiles) per block.
__global__ void proj_wmma_kernel(const float* __restrict__ A,     // [512,2048]
                                 const float* __restrict__ W,     // [512,2048]
                                 const float* __restrict__ bias,  // [512]
                                 float* __restrict__ D) {         // [512,512]
    const int lane = threadIdx.x;                  // 0..31 (full wave -> EXEC all 1s)
    const int half = lane >> 4;
    const int mn   = lane & 15;
    const int m0   = (blockIdx.y * blockDim.y + threadIdx.y) << 4;
    const int n0   = blockIdx.x << 4;

    const float* arow = A + (size_t)(m0 + mn) * IN_C + (half << 1);
    const float* brow = W + (size_t)(n0 + mn) * IN_C + (half << 1);

    v8f c = {};
    #pragma unroll 8
    for (int k = 0; k < IN_C; k += 4) {
        v2f a = *(const v2f*)(arow + k);
        v2f b = *(const v2f*)(brow + k);
        c = __builtin_amdgcn_wmma_f32_16x16x4_f32(false, a, false, b,
                                                  (short)0, c, false, false);
    }

    const int n  = n0 + mn;
    const float bv = bias[n];
    #pragma unroll
    for (int v = 0; v < 8; ++v) {
        const int m = m0 + (half << 3) + v;
        D[(size_t)m * OUT_C + n] = c[v] + bv;
    }
}

// Kernel 3: per-row L2 normalize + dist.  One 256-thread block per row (512 cols).
__global__ void normalize_kernel(const float* __restrict__ G,   // [512,512]
                                 float* __restrict__ GN,        // [512,512]
                                 float* __restrict__ dist) {    // [512]
    __shared__ float red[256];
    const int row = blockIdx.x;
    const int tid = threadIdx.x;

    const float x0 = G[(size_t)row * OUT_C + tid];
    const float x1 = G[(size_t)row * OUT_C + tid + 256];
    red[tid] = x0 * x0 + x1 * x1;
    __syncthreads();
    #pragma unroll
    for (int s = 128; s > 0; s >>= 1) {
        if (tid < s) red[tid] += red[tid + s];
        __syncthreads();
    }
    const float inv = 1.0f / fmaxf(sqrtf(red[0]), 1e-12f);   // NORM_EPS
    __syncthreads();

    const float y0 = x0 * inv, y1 = x1 * inv;
    GN[(size_t)row * OUT_C + tid]       = y0;
    GN[(size_t)row * OUT_C + tid + 256] = y1;

    const float d0 = y0 + 1e-6f, d1 = y1 + 1e-6f;            // PD_EPS
    red[tid] = d0 * d0 + d1 * d1;
    __syncthreads();
    #pragma unroll
    for (int s = 128; s > 0; s >>= 1) {
        if (tid < s) red[tid] += red[tid + s];
        __syncthreads();
    }
    if (tid == 0) dist[row] = sqrtf(red[0]);
}

// Kernel 4: out[m][n] = (sum_k gn[m][k]*gn[n][k]) / max(max(dist[m],dist[n]),1e-8)
// M=N=K=512. Same WMMA tiling; epilogue fuses the deno division.
__global__ void sim_wmma_kernel(const float* __restrict__ GN,    // [512,512]
                                const float* __restrict__ dist,  // [512]
                                float* __restrict__ out) {       // [512,512]
    const int lane = threadIdx.x;
    const int half = lane >> 4;
    const int mn   = lane & 15;
    const int m0   = (blockIdx.y * blockDim.y + threadIdx.y) << 4;
    const int n0   = blockIdx.x << 4;

    const float* arow = GN + (size_t)(m0 + mn) * OUT_C + (half << 1);
    const float* brow = GN + (size_t)(n0 + mn) * OUT_C + (half << 1);

    v8f c = {};
    #pragma unroll 8
    for (int k = 0; k < OUT_C; k += 4) {
        v2f a = *(const v2f*)(arow + k);
        v2f b = *(const v2f*)(brow + k);
        c = __builtin_amdgcn_wmma_f32_16x16x4_f32(false, a, false, b,
                                                  (short)0, c, false, false);
    }

    const int n   = n0 + mn;
    const float dj = dist[n];
    #pragma unroll
    for (int v = 0; v < 8; ++v) {
        const int m   = m0 + (half << 3) + v;
        const float di = dist[m];
        const float deno = fmaxf(fmaxf(di, dj), 1e-8f);      // DENO_EPS
        out[(size_t)m * BATCH + n] = c[v] / deno;
    }
}

extern "C" void kernel_launch(void* const* d_in, const int* in_sizes, int n_in,
                              void* d_out, int out_size, void* d_ws, size_t ws_size,
                              hipStream_t stream) {
    const float* feat = (const float*)d_in[0];   // [512,2048,196]
    const float* W    = (const float*)d_in[1];   // [512,2048]
    const float* bias = (const float*)d_in[2];   // [512]
    float* out = (float*)d_out;                  // [512,512]

    char* ws = (char*)d_ws;
    float* gpool = (float*)ws;                                              // 4 MB
    float* gproj = (float*)(ws + (size_t)BATCH * IN_C * 4);                 // 1 MB
    float* gnorm = (float*)(ws + (size_t)BATCH * IN_C * 4
                               + (size_t)BATCH * OUT_C * 4);                // 1 MB
    float* dist  = (float*)(ws + (size_t)BATCH * IN_C * 4
                               + 2 * (size_t)BATCH * OUT_C * 4);            // 2 KB

    // 1) mean pool: 1,048,576 rows, 8 waves (256 thr) per block
    const int rows = BATCH * IN_C;
    pool_mean_kernel<<<rows / 8, 256, 0, stream>>>(feat, gpool);

    // 2) projection GEMM + bias: 32x32 tiles of 16x16; 4 m-tiles per block
    proj_wmma_kernel<<<dim3(OUT_C / 16, BATCH / 64), dim3(32, 4), 0, stream>>>(
        gpool, W, bias, gproj);

    // 3) row normalize + dist
    normalize_kernel<<<BATCH, 256, 0, stream>>>(gproj, gnorm, dist);

    // 4) cosine-sim GEMM fused with deno division
    sim_wmma_kernel<<<dim3(BATCH / 16, BATCH / 64), dim3(32, 4), 0, stream>>>(
        gnorm, dist, out);
}